// PerExampleParamsNet_30039001268703
// MI455X (gfx1250) — compile-verified
//
#include <hip/hip_runtime.h>
#include <hip/hip_bf16.h>

typedef __attribute__((ext_vector_type(2))) float v2f;
typedef __attribute__((ext_vector_type(8))) float v8f;

namespace {
constexpr int kB    = 128;
constexpr int kDin  = 512;
constexpr int kDh   = 1024;
constexpr int kDout = 512;
constexpr int kThreads = 256;          // 8 wave32s per block
constexpr int kWaves   = kThreads / 32;
}

// Per-wave mat-vec tile via V_WMMA_F32_16X16X4_F32.
//
// A (16x4 f32) lane layout per ISA: lanes 0-15 hold rows M=0..15 with
// VGPR0=K0, VGPR1=K1; lanes 16-31 hold the same rows with VGPR0=K2,
// VGPR1=K3.  A single per-lane 8-byte load at
//   W + (m0 + lane%16)*ldk + k + 2*(lane/16)
// lands in exactly that layout -- no permutes needed.
//
// B (4x16 f32) is filled with the vector broadcast across all 16 columns
// (B[k,n] = v[k] for all n), so every column of D holds the same dot
// product: the matrix unit performs the K-reduction across lanes for us.

__global__ __launch_bounds__(kThreads)
void per_ex_layer1(const float* __restrict__ x,
                   const float* __restrict__ W1,
                   const float* __restrict__ b1,
                   float* __restrict__ h) {
  __shared__ float sx[kDin];
  const int b   = blockIdx.y;
  const int tid = threadIdx.x;

  // Stage x[b] (2 KB) into LDS, coalesced.
  for (int i = tid; i < kDin; i += kThreads) sx[i] = x[b * kDin + i];
  __syncthreads();

  const int lane = tid & 31;
  const int wave = tid >> 5;
  const int r    = lane & 15;     // row within tile
  const int half = lane >> 4;     // 0: K0/K1  1: K2/K3
  const int m0   = (blockIdx.x * kWaves + wave) * 16;

  const float* __restrict__ row =
      W1 + (size_t)b * kDh * kDin + (size_t)(m0 + r) * kDin + half * 2;

  v8f acc = {};
  #pragma unroll 8
  for (int k = 0; k < kDin; k += 4) {
    // Stream-once weights: non-temporal so 537 MB doesn't thrash 192 MB L2.
    v2f a  = __builtin_nontemporal_load((const v2f*)(row + k));
    v2f bv = *(const v2f*)(sx + k + half * 2);   // x[k..k+1] or x[k+2..k+3]
    acc = __builtin_amdgcn_wmma_f32_16x16x4_f32(
        /*neg_a=*/false, a, /*neg_b=*/false, bv,
        /*c_mod=*/(short)0, acc, /*reuse_a=*/false, /*reuse_b=*/false);
  }

  // D layout: VGPR j = rows M=j (lanes 0-15) and M=8+j (lanes 16-31); all 16
  // columns identical, so lanes 0 and 16 carry the full tile.
  if (r == 0) {
    const int mo = m0 + half * 8;
    #pragma unroll
    for (int j = 0; j < 8; ++j) {
      float v = acc[j] + b1[b * kDh + mo + j];
      h[b * kDh + mo + j] = v > 0.0f ? v : 0.0f;   // fused ReLU
    }
  }
}

__global__ __launch_bounds__(kThreads)
void per_ex_layer2(const float* __restrict__ W2,
                   const float* __restrict__ b2,
                   const float* __restrict__ h,
                   float* __restrict__ out) {
  __shared__ float sh[kDh];
  const int b   = blockIdx.y;
  const int tid = threadIdx.x;

  for (int i = tid; i < kDh; i += kThreads) sh[i] = h[b * kDh + i];
  __syncthreads();

  const int lane = tid & 31;
  const int wave = tid >> 5;
  const int r    = lane & 15;
  const int half = lane >> 4;
  const int m0   = (blockIdx.x * kWaves + wave) * 16;

  const float* __restrict__ row =
      W2 + (size_t)b * kDout * kDh + (size_t)(m0 + r) * kDh + half * 2;

  v8f acc = {};
  #pragma unroll 8
  for (int k = 0; k < kDh; k += 4) {
    v2f a  = __builtin_nontemporal_load((const v2f*)(row + k));
    v2f bv = *(const v2f*)(sh + k + half * 2);
    acc = __builtin_amdgcn_wmma_f32_16x16x4_f32(
        false, a, false, bv, (short)0, acc, false, false);
  }

  if (r == 0) {
    const int mo = m0 + half * 8;
    #pragma unroll
    for (int j = 0; j < 8; ++j) {
      out[b * kDout + mo + j] = acc[j] + b2[b * kDout + mo + j];
    }
  }
}

extern "C" void kernel_launch(void* const* d_in, const int* in_sizes, int n_in,
                              void* d_out, int out_size, void* d_ws, size_t ws_size,
                              hipStream_t stream) {
  const float* x  = (const float*)d_in[0];   // [128, 512]
  const float* W1 = (const float*)d_in[1];   // [128, 1024, 512]
  const float* b1 = (const float*)d_in[2];   // [128, 1024]
  const float* W2 = (const float*)d_in[3];   // [128, 512, 1024]
  const float* b2 = (const float*)d_in[4];   // [128, 512]
  float* out = (float*)d_out;                // [128, 512]
  float* h   = (float*)d_ws;                 // scratch: 128*1024 f32 = 512 KB

  // Layer 1: 1024 row-tiles of 16 -> 64 tiles/example, 8 waves/block
  dim3 g1(kDh / 16 / kWaves, kB);            // (8, 128)
  per_ex_layer1<<<g1, kThreads, 0, stream>>>(x, W1, b1, h);

  // Layer 2: 512 rows -> 32 tiles/example
  dim3 g2(kDout / 16 / kWaves, kB);          // (4, 128)
  per_ex_layer2<<<g2, kThreads, 0, stream>>>(W2, b2, h, out);
}